// Bidct_44976897524646
// MI455X (gfx1250) — compile-verified
//
#include <hip/hip_runtime.h>

// CDNA5 (gfx1250) fp32 WMMA types
typedef __attribute__((ext_vector_type(2))) float v2f;   // A/B operand of 16x16x4 f32 (2 VGPRs)
typedef __attribute__((ext_vector_type(8))) float v8f;   // C/D 16x16 f32 (8 VGPRs)

// D = A(16x4) x B(4x16) + C, fp32, 8-arg form: (neg_a, A, neg_b, B, c_mod, C, reuse_a, reuse_b)
#define WMMA4(A, B, C) \
  __builtin_amdgcn_wmma_f32_16x16x4_f32(false, (A), false, (B), (short)0, (C), false, false)

// One wave == one 16x16 image tile == four 8x8 JPEG blocks (blockdiag(M,M) trick).
__global__ __launch_bounds__(128) void idct8x8_wmma_kernel(
    const float* __restrict__ x,
    const float* __restrict__ qtab,
    const float* __restrict__ mtx,
    float* __restrict__ out)
{
  const int lane = threadIdx.x & 31;
  const int wave = threadIdx.x >> 5;
  const int tile = blockIdx.x * 4 + wave;      // global tile id
  const int b    = tile >> 12;                 // 64*64 = 4096 tiles per 1024x1024 image
  const int rem  = tile & 4095;
  const int tr   = rem >> 6;                   // tile row 0..63
  const int tc   = rem & 63;                   // tile col 0..63

  const int hf  = lane >> 4;                   // half-wave: 0 = lanes 0-15, 1 = lanes 16-31
  const int l16 = lane & 15;
  const int cb  = hf << 1;                     // K sub-offset handled by this half-wave (0 or 2)

  const size_t base = ((size_t)b << 20) + ((size_t)(tr << 4) << 10) + (size_t)(tc << 4);

  // ---- E = blockdiag(M, M): identical per-lane contents as pass-1 B-chunks and
  //      pass-2 A-chunks under the ISA 16x4-A / 4x16-B fp32 layouts ----
  v2f e[4];
  {
    const int cblk = (l16 < 8) ? l16 : (l16 - 8);
#pragma unroll
    for (int k = 0; k < 4; ++k) {
      const int  r0    = 4 * k + cb;                    // E rows r0, r0+1 for this lane
      const int  rblk  = (r0 < 8) ? r0 : (r0 - 8);
      const bool valid = (r0 < 8) == (l16 < 8);         // on-diagonal block?
      const float mx = mtx[rblk * 8 + cblk];            // always-valid address, masked by select
      const float my = mtx[(rblk + 1) * 8 + cblk];
      e[k].x = valid ? mx : 0.0f;
      e[k].y = valid ? my : 0.0f;
    }
  }

  // ---- per-lane dequant constants: q row = lane%8, columns {cb,cb+1,cb+4,cb+5} ----
  const int qrow = lane & 7;
  const float q0 = qtab[qrow * 8 + cb + 0];
  const float q1 = qtab[qrow * 8 + cb + 1];
  const float q4 = qtab[qrow * 8 + cb + 4];
  const float q5 = qtab[qrow * 8 + cb + 5];

  // ---- load tile rows in A-operand layout: lane row = l16, cols k*4 + cb + {0,1} ----
  const float* rowp = x + base + ((size_t)l16 << 10) + cb;
  v2f a0 = *(const v2f*)(rowp + 0);
  v2f a1 = *(const v2f*)(rowp + 4);
  v2f a2 = *(const v2f*)(rowp + 8);
  v2f a3 = *(const v2f*)(rowp + 12);

  // dequantize (qtable repeats every 8 columns -> chunks 0,2 share q0/q1; 1,3 share q4/q5)
  a0.x *= q0; a0.y *= q1;
  a1.x *= q4; a1.y *= q5;
  a2.x *= q0; a2.y *= q1;
  a3.x *= q4; a3.y *= q5;

  // ---- pass 1: T = (Q .* Y) x E, K=16 as four chained 16x16x4 WMMAs ----
  v8f t = {0.f, 0.f, 0.f, 0.f, 0.f, 0.f, 0.f, 0.f};
  t = WMMA4(a0, e[0], t);
  t = WMMA4(a1, e[1], t);
  t = WMMA4(a2, e[2], t);
  t = WMMA4(a3, e[3], t);

  // ---- re-layout T (C/D layout) into pass-2 B operands: half of each chunk lives in
  //      the other half-wave -> 8 half-wave swaps + branch-free selects ----
  float ts[8];
#pragma unroll
  for (int p = 0; p < 8; ++p) ts[p] = __shfl_xor(t[p], 16, 32);

  const bool lo = (hf == 0);
  v2f b0 = { lo ? t[0]  : ts[2], lo ? t[1]  : ts[3] };  // K rows 0..3
  v2f b1 = { lo ? t[4]  : ts[6], lo ? t[5]  : ts[7] };  // K rows 4..7
  v2f b2 = { lo ? ts[0] : t[2],  lo ? ts[1] : t[3]  };  // K rows 8..11
  v2f b3 = { lo ? ts[4] : t[6],  lo ? ts[5] : t[7]  };  // K rows 12..15

  // ---- pass 2: Out = E^T x T + 128 (the +128 rides in on the C operand) ----
  v8f d = {128.f, 128.f, 128.f, 128.f, 128.f, 128.f, 128.f, 128.f};
  d = WMMA4(e[0], b0, d);
  d = WMMA4(e[1], b1, d);
  d = WMMA4(e[2], b2, d);
  d = WMMA4(e[3], b3, d);

  // ---- store straight from C/D layout: row = p + 8*hf, col = l16 ----
  float* orow = out + base + ((size_t)(hf << 3) << 10) + l16;
#pragma unroll
  for (int p = 0; p < 8; ++p) orow[(size_t)p << 10] = d[p];
}

extern "C" void kernel_launch(void* const* d_in, const int* in_sizes, int n_in,
                              void* d_out, int out_size, void* d_ws, size_t ws_size,
                              hipStream_t stream) {
  const float* x    = (const float*)d_in[0];   // (32,1,1024,1024) fp32
  const float* qtab = (const float*)d_in[1];   // (8,8) fp32
  const float* mtx  = (const float*)d_in[2];   // (8,8) fp32
  float* o = (float*)d_out;

  const int total  = in_sizes[0];      // 32 * 1024 * 1024
  const int tiles  = total / 256;      // 16x16 tiles -> 131072
  const int blocks = tiles / 4;        // 4 waves (128 threads) per workgroup

  idct8x8_wmma_kernel<<<blocks, 128, 0, stream>>>(x, qtab, mtx, o);
}